// CHKGAT_35450660061923
// MI455X (gfx1250) — compile-verified
//
#include <hip/hip_runtime.h>
#include <cmath>
#include <stdint.h>

#define DIM       128
#define BATCH     64
#define NUM_ITEM  16384
#define NUM_REL   16
#define PAD       132      // LDS row stride in floats: 132%64==4 -> conflict-free row access
#define ITILE     32       // items per block

typedef float v2f __attribute__((ext_vector_type(2)));
typedef float v8f __attribute__((ext_vector_type(8)));

// CDNA5 async global->LDS copy (16B per lane), tracked by ASYNCcnt.
// VDST takes the wave-relative LDS byte offset == low 32 bits of the generic
// pointer to a __shared__ object (flat LDS aperture keeps offset in addr[31:0]).
__device__ __forceinline__ void async_copy_b128(void* lds_dst, const void* gsrc) {
    uint32_t lds_off = (uint32_t)(uintptr_t)lds_dst;
    asm volatile("global_load_async_to_lds_b128 %0, %1, off"
                 :: "v"(lds_off), "v"(gsrc) : "memory");
}
__device__ __forceinline__ void async_wait_all() {
    asm volatile("s_wait_asynccnt 0" ::: "memory");
}

// Kernel A: gather user rows, pre-add "buy", compute predict[b] (64 sigmoids).
// ws layout: [0, 8192)   = ue[b][d]  (user embeds)
//            [8192,16384) = su[b][d] (user + buy)
__global__ __launch_bounds__(DIM) void gather_predict_kernel(
    const int* __restrict__ users, const int* __restrict__ items,
    const float* __restrict__ ent, const float* __restrict__ rel,
    float* __restrict__ out, float* __restrict__ ws)
{
    const int b = blockIdx.x;
    const int d = threadIdx.x;
    const int u  = users[b];
    const int it = items[b];
    const float uv  = ent[(long)u  * DIM + d];
    const float iv  = ent[(long)it * DIM + d];
    const float buy = rel[(NUM_REL - 1) * DIM + d];
    const float su  = uv + buy;
    ws[b * DIM + d] = uv;
    ws[BATCH * DIM + b * DIM + d] = su;

    __shared__ float r1[DIM], r2[DIM];
    r1[d] = uv * iv;
    r2[d] = fabsf(su - iv);
    __syncthreads();
    for (int s = DIM / 2; s > 0; s >>= 1) {
        if (d < s) { r1[d] += r1[d + s]; r2[d] += r2[d + s]; }
        __syncthreads();
    }
    if (d == 0) {
        float x = r1[0] + r2[0];
        out[b] = 1.0f / (1.0f + __expf(-x));
    }
}

// Kernel B: ranking_predict. One block = 32 items x all 64 batch rows.
// 8 waves, each wave owns a 16x16 C tile: score via V_WMMA_F32_16X16X4_F32,
// L1 distance via VALU in the same C-matrix register layout.
__global__ __launch_bounds__(256) void ranking_kernel(
    const float* __restrict__ ent, const float* __restrict__ ws,
    float* __restrict__ out)
{
    __shared__ float su_s[BATCH * PAD];   // shifted user rows, padded
    __shared__ float it_s[ITILE * PAD];   // item tile rows, padded

    const int tid = threadIdx.x;
    const int item_base = blockIdx.x * ITILE;
    const float* ue_ws = ws;                  // BATCH x DIM (A matrix source)
    const float* su_ws = ws + BATCH * DIM;    // BATCH x DIM shifted user rows

    // Async-stage su (2048 float4) and item tile (1024 float4) into padded LDS.
#pragma unroll
    for (int q = tid; q < BATCH * (DIM / 4); q += 256) {
        int row = q >> 5;               // 32 float4 per row
        int c4  = (q & 31) << 2;
        async_copy_b128(su_s + row * PAD + c4, su_ws + row * DIM + c4);
    }
#pragma unroll
    for (int q = tid; q < ITILE * (DIM / 4); q += 256) {
        int row = q >> 5;
        int c4  = (q & 31) << 2;
        async_copy_b128(it_s + row * PAD + c4,
                        ent + (long)(item_base + row) * DIM + c4);
    }
    async_wait_all();     // this wave's async copies complete
    __syncthreads();      // all waves' tiles visible

    const int wave = tid >> 5;
    const int lane = tid & 31;
    const int ln   = lane & 15;
    const int g    = lane >> 4;    // lane half: 0 or 1
    const int mb   = wave & 3;     // batch tile index  (rows mb*16..)
    const int nb   = wave >> 2;    // item tile index   (cols nb*16..)

    // --- Score: C(16x16) = sum_k A(16x4) * B(4x16), K = 128 in 32 steps.
    // A layout (ISA 32-bit A 16x4): lane holds A[m=ln][k = 2*g + vgpr] -> float2.
    // B layout mirrors it: lane holds B[k = 2*g + vgpr][n=ln] -> contiguous pair of row n.
    v8f acc = {};
    const float* arow = ue_ws + (mb * 16 + ln) * DIM + 2 * g;       // global (L2-hot)
    const float* brow = it_s  + (nb * 16 + ln) * PAD + 2 * g;       // LDS, padded
#pragma unroll
    for (int kk = 0; kk < 32; ++kk) {
        v2f a = *(const v2f*)(arow + kk * 4);
        v2f b = *(const v2f*)(brow + kk * 4);
        acc = __builtin_amdgcn_wmma_f32_16x16x4_f32(
            /*neg_a=*/false, a, /*neg_b=*/false, b,
            /*c_mod=*/(short)0, acc, /*reuse_a=*/false, /*reuse_b=*/false);
    }

    // --- L1 distance in the same C layout: VGPR v holds row m = v + 8*g, col n = ln.
    float dist[8] = {0.f,0.f,0.f,0.f,0.f,0.f,0.f,0.f};
    const float* itrow  = it_s + (nb * 16 + ln) * PAD;
    const float* subase = su_s + (mb * 16 + 8 * g) * PAD;
#pragma unroll 2
    for (int d0 = 0; d0 < DIM; d0 += 4) {
        float4 t = *(const float4*)(itrow + d0);
#pragma unroll
        for (int v = 0; v < 8; ++v) {
            float4 s = *(const float4*)(subase + v * PAD + d0);
            dist[v] += fabsf(s.x - t.x) + fabsf(s.y - t.y)
                     + fabsf(s.z - t.z) + fabsf(s.w - t.w);
        }
    }

    // --- Fuse: sigmoid(dist + score), store. predict occupies out[0..63].
    float* rank = out + BATCH;
    const int col = item_base + nb * 16 + ln;
#pragma unroll
    for (int v = 0; v < 8; ++v) {
        int row = mb * 16 + 8 * g + v;
        float x = acc[v] + dist[v];
        rank[(long)row * NUM_ITEM + col] = 1.0f / (1.0f + __expf(-x));
    }
}

extern "C" void kernel_launch(void* const* d_in, const int* in_sizes, int n_in,
                              void* d_out, int out_size, void* d_ws, size_t ws_size,
                              hipStream_t stream) {
    const int*   users = (const int*)d_in[0];
    const int*   items = (const int*)d_in[1];
    const float* ent   = (const float*)d_in[2];
    const float* rel   = (const float*)d_in[3];
    float* out = (float*)d_out;
    float* ws  = (float*)d_ws;   // needs 16384 floats = 64 KB

    gather_predict_kernel<<<BATCH, DIM, 0, stream>>>(users, items, ent, rel, out, ws);
    ranking_kernel<<<NUM_ITEM / ITILE, 256, 0, stream>>>(ent, ws, out);
}